// LazyMLPBlock_45612552684061
// MI455X (gfx1250) — compile-verified
//
#include <hip/hip_runtime.h>
#include <hip/hip_bf16.h>
#include <math.h>

#define SEQ   16
#define HDIM  1024
#define IDIM  1024
#define NE    32
#define TOPK  4
#define TWO_I 2048
#define FLIMIT 7.0f
#define FALPHA 1.702f

typedef float v2f __attribute__((ext_vector_type(2)));
typedef float v4f __attribute__((ext_vector_type(4)));
typedef float v8f __attribute__((ext_vector_type(8)));

__device__ __forceinline__ v8f wmma_f32_16x16x4(v2f a, v2f b, v8f c) {
  // D = A(16x4 f32) * B(4x16 f32) + C(16x16 f32), wave32, one matrix per wave
  return __builtin_amdgcn_wmma_f32_16x16x4_f32(false, a, false, b, (short)0, c,
                                               false, false);
}

__device__ __forceinline__ float swiglu_pair(float hg, float hl) {
  float xg = fminf(hg, FLIMIT);
  float xl = fminf(fmaxf(hl, -FLIMIT), FLIMIT);
  float sg = 1.0f / (1.0f + __expf(-FALPHA * xg));
  return xg * sg * (xl + 1.0f);
}

// ---------------------------------------------------------------- kernel 1
// RMSNorm + gate logits + top-k + softmax + per-expert token lists.
__global__ __launch_bounds__(256) void k1_gate(
    const float* __restrict__ x, const float* __restrict__ nscale,
    const float* __restrict__ gate_w, float* __restrict__ t_ws,
    float* __restrict__ ew_ws, int* __restrict__ slot_of,
    int* __restrict__ tok_of, int* __restrict__ count) {
  __shared__ float red[256];
  __shared__ float g[SEQ][NE];
  __shared__ int sidx[SEQ][TOPK];
  __shared__ float inv_s;
  const int tid = threadIdx.x;

  for (int s = 0; s < SEQ; ++s) {
    float p = 0.f;
    for (int i = tid; i < HDIM; i += 256) {
      float v = x[s * HDIM + i];
      p += v * v;
    }
    red[tid] = p;
    __syncthreads();
    for (int off = 128; off > 0; off >>= 1) {
      if (tid < off) red[tid] += red[tid + off];
      __syncthreads();
    }
    if (tid == 0) inv_s = rsqrtf(red[0] / (float)HDIM + 1e-5f);
    __syncthreads();
    const float inv = inv_s;
    for (int i = tid; i < HDIM; i += 256)
      t_ws[s * HDIM + i] = x[s * HDIM + i] * inv * nscale[i];
    __syncthreads();
  }

  // gate logits: 512 dot products of length 1024
  for (int p = tid; p < SEQ * NE; p += 256) {
    const int s = p >> 5, e = p & 31;
    const float* tr = t_ws + s * HDIM;
    const float* wr = gate_w + e * HDIM;
    float acc = 0.f;
    for (int i = 0; i < HDIM; ++i) acc += tr[i] * wr[i];
    g[s][e] = acc;
  }
  __syncthreads();

  if (tid < SEQ) {
    const int s = tid;
    unsigned used = 0u;
    float v[TOPK];
    for (int k = 0; k < TOPK; ++k) {
      float best = -3.4e38f;
      int bi = 0;
      for (int e = 0; e < NE; ++e)
        if (!((used >> e) & 1u) && g[s][e] > best) { best = g[s][e]; bi = e; }
      used |= (1u << bi);
      v[k] = best;
      sidx[s][k] = bi;
    }
    const float m = v[0];  // sorted descending
    float ex[TOPK], sum = 0.f;
    for (int k = 0; k < TOPK; ++k) { ex[k] = __expf(v[k] - m); sum += ex[k]; }
    for (int k = 0; k < TOPK; ++k) ew_ws[s * TOPK + k] = ex[k] / sum;
  }
  __syncthreads();

  if (tid == 0) {
    int cnt[NE];
    for (int e = 0; e < NE; ++e) cnt[e] = 0;
    for (int s = 0; s < SEQ; ++s)
      for (int k = 0; k < TOPK; ++k) {
        const int e = sidx[s][k];
        const int pos = cnt[e]++;
        slot_of[s * TOPK + k] = e * 16 + pos;
        tok_of[e * 16 + pos] = s;
      }
    for (int e = 0; e < NE; ++e) count[e] = cnt[e];
  }
}

// Shared WMMA streaming loop: A = two weight-row tiles (NT), B = one
// gathered-activation tile; outer chunk loop hoists the prefetch so the
// inner body is pure load+wmma.
__device__ __forceinline__ void gemm_stream(const float* __restrict__ rowA0,
                                            const float* __restrict__ rowA1,
                                            const float* __restrict__ rowB,
                                            int kdim, int koff, v8f& acc0,
                                            v8f& acc1) {
  for (int kb = 0; kb < kdim; kb += 64) {
    if (kb + 64 < kdim) {  // uniform scalar test, once per 16 WMMA pairs
      __builtin_prefetch(rowA0 + kb + 64, 0, 1);
      __builtin_prefetch(rowA1 + kb + 64, 0, 1);
    }
#pragma unroll
    for (int ku = 0; ku < 64; ku += 4) {
      const int k = kb + ku;
      v2f a0 = __builtin_nontemporal_load((const v2f*)(rowA0 + k + koff));
      v2f a1 = __builtin_nontemporal_load((const v2f*)(rowA1 + k + koff));
      v2f b = *(const v2f*)(rowB + k + koff);
      acc0 = wmma_f32_16x16x4(a0, b, acc0);
      acc1 = wmma_f32_16x16x4(a1, b, acc1);
    }
  }
}

// ---------------------------------------------------------------- kernel 2
// Per expert: h = W1 * T^ + b1 (WMMA, A = weight rows, B = gathered tokens),
// fused interleaved SwiGLU, write activations (16 slots x IDIM).
__global__ __launch_bounds__(256) void k2_mlp1(
    const float* __restrict__ w1, const float* __restrict__ b1,
    const float* __restrict__ t_ws, const int* __restrict__ tok_of,
    const int* __restrict__ count, float* __restrict__ a_buf) {
  const int e = blockIdx.x;
  const int cnt = count[e];
  if (cnt == 0) return;  // unselected experts: skip their 12MB of weights
  const int tid = threadIdx.x;
  const int wave = tid >> 5;
  const int lane = tid & 31;
  const int half = lane >> 4;
  const int n = lane & 15;
  const int col0 = (blockIdx.y * 8 + wave) * 32;  // two 16-col tiles per wave
  const int koff = half * 2;                      // A/B lane K split

  const float* w1e = w1 + (size_t)e * TWO_I * HDIM;
  const float* rowA0 = w1e + (size_t)(col0 + n) * HDIM;
  const float* rowA1 = w1e + (size_t)(col0 + 16 + n) * HDIM;
  const int tok = (n < cnt) ? tok_of[e * 16 + n] : 0;
  const float* rowB = t_ws + tok * HDIM;

  v8f acc0 = {0.f, 0.f, 0.f, 0.f, 0.f, 0.f, 0.f, 0.f};
  v8f acc1 = acc0;
  gemm_stream(rowA0, rowA1, rowB, HDIM, koff, acc0, acc1);

  const float* b1e = b1 + e * TWO_I;
  float c0[8], c1[8];
#pragma unroll
  for (int r = 0; r < 8; ++r) {
    c0[r] = acc0[r] + b1e[col0 + 8 * half + r];
    c1[r] = acc1[r] + b1e[col0 + 16 + 8 * half + r];
  }
  v4f act0, act1;
#pragma unroll
  for (int j = 0; j < 4; ++j) {
    act0[j] = swiglu_pair(c0[2 * j], c0[2 * j + 1]);
    act1[j] = swiglu_pair(c1[2 * j], c1[2 * j + 1]);
  }
  float* arow = a_buf + (size_t)(e * 16 + n) * IDIM;
  const int ac = col0 / 2 + 4 * half;
  *(v4f*)(arow + ac) = act0;
  *(v4f*)(arow + ac + 8) = act1;
}

// ---------------------------------------------------------------- kernel 3
// Per expert: o = W2 * A^ + b2 (WMMA), store per-slot outputs (16 x HDIM).
__global__ __launch_bounds__(256) void k3_mlp2(
    const float* __restrict__ w2, const float* __restrict__ b2,
    const float* __restrict__ a_buf, const int* __restrict__ count,
    float* __restrict__ o_buf) {
  const int e = blockIdx.x;
  const int cnt = count[e];
  if (cnt == 0) return;
  const int tid = threadIdx.x;
  const int wave = tid >> 5;
  const int lane = tid & 31;
  const int half = lane >> 4;
  const int n = lane & 15;
  const int col0 = (blockIdx.y * 8 + wave) * 32;
  const int koff = half * 2;

  const float* w2e = w2 + (size_t)e * HDIM * IDIM;
  const float* rowA0 = w2e + (size_t)(col0 + n) * IDIM;
  const float* rowA1 = w2e + (size_t)(col0 + 16 + n) * IDIM;
  const float* rowB = a_buf + (size_t)(e * 16 + n) * IDIM;

  v8f acc0 = {0.f, 0.f, 0.f, 0.f, 0.f, 0.f, 0.f, 0.f};
  v8f acc1 = acc0;
  gemm_stream(rowA0, rowA1, rowB, IDIM, koff, acc0, acc1);

  const float* b2e = b2 + e * HDIM;
  float c0[8], c1[8];
#pragma unroll
  for (int r = 0; r < 8; ++r) {
    c0[r] = acc0[r] + b2e[col0 + 8 * half + r];
    c1[r] = acc1[r] + b2e[col0 + 16 + 8 * half + r];
  }
  float* orow = o_buf + (size_t)(e * 16 + n) * HDIM;
  const int oc = col0 + 8 * half;
  v4f lo0 = {c0[0], c0[1], c0[2], c0[3]};
  v4f hi0 = {c0[4], c0[5], c0[6], c0[7]};
  v4f lo1 = {c1[0], c1[1], c1[2], c1[3]};
  v4f hi1 = {c1[4], c1[5], c1[6], c1[7]};
  *(v4f*)(orow + oc) = lo0;
  *(v4f*)(orow + oc + 4) = hi0;
  *(v4f*)(orow + oc + 16) = lo1;
  *(v4f*)(orow + oc + 20) = hi1;
}

// ---------------------------------------------------------------- kernel 4
// out = x + sum_k ew[s,k] * o_buf[slot(s,k)]
__global__ __launch_bounds__(256) void k4_combine(
    const float* __restrict__ x, const float* __restrict__ ew_ws,
    const int* __restrict__ slot_of, const float* __restrict__ o_buf,
    float* __restrict__ out) {
  const int gid = blockIdx.x * 256 + threadIdx.x;
  const int s = gid >> 10;
  const int c = gid & 1023;
  float acc = x[gid];
#pragma unroll
  for (int k = 0; k < TOPK; ++k) {
    const int slot = slot_of[s * TOPK + k];
    acc += ew_ws[s * TOPK + k] * o_buf[(size_t)slot * HDIM + c];
  }
  out[gid] = acc;
}

extern "C" void kernel_launch(void* const* d_in, const int* in_sizes, int n_in,
                              void* d_out, int out_size, void* d_ws,
                              size_t ws_size, hipStream_t stream) {
  (void)in_sizes; (void)n_in; (void)out_size; (void)ws_size;
  const float* x = (const float*)d_in[0];
  const float* nscale = (const float*)d_in[1];
  const float* gate_w = (const float*)d_in[2];
  const float* w1 = (const float*)d_in[3];
  const float* b1 = (const float*)d_in[4];
  const float* w2 = (const float*)d_in[5];
  const float* b2 = (const float*)d_in[6];
  float* out = (float*)d_out;

  // workspace layout (~4.3 MB)
  float* t_ws = (float*)d_ws;                      // 16K floats
  float* ew_ws = t_ws + SEQ * HDIM;                // 64 floats
  int* slot_of = (int*)(ew_ws + SEQ * TOPK);       // 64 ints
  int* tok_of = slot_of + SEQ * TOPK;              // 512 ints
  int* count = tok_of + NE * 16;                   // 32 ints
  float* a_buf = (float*)((char*)d_ws + (1 << 17));  // 512 x 1024 floats
  float* o_buf = a_buf + (size_t)NE * 16 * IDIM;     // 512 x 1024 floats

  k1_gate<<<1, 256, 0, stream>>>(x, nscale, gate_w, t_ws, ew_ws, slot_of,
                                 tok_of, count);
  k2_mlp1<<<dim3(NE, 8), 256, 0, stream>>>(w1, b1, t_ws, tok_of, count, a_buf);
  k3_mlp2<<<dim3(NE, 4), 256, 0, stream>>>(w2, b2, a_buf, count, o_buf);
  k4_combine<<<64, 256, 0, stream>>>(x, ew_ws, slot_of, o_buf, out);
}